// GUAN_59811714564807
// MI455X (gfx1250) — compile-verified
//
#include <hip/hip_runtime.h>

// ---------------------------------------------------------------------------
// MI455X (gfx1250) implementation.
// Problem constants from the reference (fixed for this harness).
// ---------------------------------------------------------------------------
#define B_    4
#define N_    50000
#define E_    800000
#define XIN_  32
#define XOUT_ 32
#define EIN_  16
#define EOUT_ 32
#define KD_   (EIN_ + 2 * XIN_)   // 80 = concat feature depth
#define FSTRIDE 84                 // padded LDS row stride (floats): 336 B is
                                   // 16B-aligned for b128 stores, low-conflict

typedef float v2f __attribute__((ext_vector_type(2)));
typedef float v4f __attribute__((ext_vector_type(4)));
typedef float v8f __attribute__((ext_vector_type(8)));

// ---------------------------------------------------------------------------
// Zero workspace (sums: B*N floats, counts: N floats) — required every call
// since atomics accumulate and the harness does not re-poison between replays.
// ---------------------------------------------------------------------------
__global__ void zero_ws_kernel(float* __restrict__ p, int n) {
    int i = blockIdx.x * blockDim.x + threadIdx.x;
    if (i < n) p[i] = 0.0f;
}

// ---------------------------------------------------------------------------
// counts[n] = number of edges with dst == n  (batch independent)
// ---------------------------------------------------------------------------
__global__ void counts_kernel(const int* __restrict__ edges, float* __restrict__ counts) {
    int i = blockIdx.x * blockDim.x + threadIdx.x;
    if (i < E_) atomicAdd(&counts[edges[2 * i + 1]], 1.0f);
}

// ---------------------------------------------------------------------------
// Edge MLP: new_e = relu([e | x[src] | x[dst]] @ W1 + b1), fused per-edge mean
// scatter-add into sums[b*N + dst].
//
// One wave32 per tile of 32 edges (one batch):
//   - lane l owns edge e0+l (indices loaded once, used for gather + scatter)
//   - M=32 as two 16-row A-tiles, N=32 as two 16-col B-tiles
//   - 20 k-steps x 4 WMMAs = 80 x v_wmma_f32_16x16x4_f32 per wave
//   - W1 pre-swizzled frag-major in LDS: each B frag is ONE ds_load_b64,
//     lane-stride-2 layout covers all 64 banks conflict-free
// ---------------------------------------------------------------------------
__global__ __launch_bounds__(256) void edge_mlp_kernel(
    const float* __restrict__ x,      // (B, N, 32)
    const float* __restrict__ e,      // (B, E, 16)
    const int*   __restrict__ edges,  // (E, 2)
    const float* __restrict__ W1,     // (80, 32) row-major
    const float* __restrict__ b1,     // (32,)
    float*       __restrict__ new_e,  // (B, E, 32)
    float*       __restrict__ sums)   // (B, N)
{
    __shared__ float W1f[20 * 2 * 32 * 2];     // frag-major [kstep][ntile][lane][2]
    __shared__ float b1s[EOUT_];
    __shared__ float feat[8 * 32 * FSTRIDE];   // 8 waves * 32 edges * 84 = 86016 B
    __shared__ float red[8 * 32 * 17];         // mean-reduction transpose scratch

    const int tid  = threadIdx.x;
    const int lane = tid & 31;
    const int wave = tid >> 5;

    // ---- Pre-swizzle W1 into per-lane fragment order (block-wide) ----
    // Lane l, k-step ks, n-tile nt needs W1[ks*4 + (l>>4)*2 + j][nt*16 + (l&15)]
    // for j in {0,1}, stored contiguously -> single aligned ds_load_b64.
    for (int i = tid; i < 20 * 2 * 32 * 2; i += 256) {
        const int j  = i & 1;
        const int l  = (i >> 1) & 31;
        const int nt = (i >> 6) & 1;
        const int ks = i >> 7;
        const int kk = ks * 4 + ((l >> 4) << 1) + j;
        const int nn = nt * 16 + (l & 15);
        W1f[i] = W1[kk * EOUT_ + nn];
    }
    if (tid < EOUT_) b1s[tid] = b1[tid];
    __syncthreads();

    const int b      = blockIdx.y;
    const int ntiles = E_ / 32;
    int  tile  = blockIdx.x * 8 + wave;
    bool valid = (tile < ntiles);              // wave-uniform
    if (!valid) tile = 0;                      // keep EXEC all-1s for WMMA
    const int e0 = tile * 32;

    // ---- Edge indices: lane l owns edge e0+l ----
    const int myedge = e0 + lane;
    const int srcl   = edges[2 * myedge + 0];
    const int dstl   = edges[2 * myedge + 1];

    // ---- Stage gathered 32x80 feature tile into LDS; lane l fills row l ----
    float* ft = &feat[wave * 32 * FSTRIDE];
    {
        const v4f* ep = (const v4f*)(e + ((size_t)b * E_ + myedge) * EIN_);
        const v4f* sp = (const v4f*)(x + ((size_t)b * N_ + srcl)   * XIN_);
        const v4f* dp = (const v4f*)(x + ((size_t)b * N_ + dstl)   * XIN_);
        v4f* lp = (v4f*)(ft + lane * FSTRIDE);
        #pragma unroll
        for (int s = 0; s < 4; ++s) lp[s]      = __builtin_nontemporal_load(ep + s);
        #pragma unroll
        for (int s = 0; s < 8; ++s) lp[4 + s]  = sp[s];   // x is L2-resident: RT
        #pragma unroll
        for (int s = 0; s < 8; ++s) lp[12 + s] = dp[s];
    }
    // Wave-internal visibility of LDS stores (gfx1250 split counter).
    asm volatile("s_wait_dscnt 0" ::: "memory");

    // ---- WMMA fragment indices (ISA 7.12.2, 32-bit 16x4 A / 4x16 B) ----
    const int half = lane >> 4;      // 0: lanes 0-15, 1: lanes 16-31
    const int mn   = lane & 15;      // row (A) / col (B,C,D) index
    const int kb   = half * 2;       // K sub-offset within a 4-deep step

    v8f acc00 = {};                  // edges  0-15, cols  0-15
    v8f acc01 = {};                  // edges  0-15, cols 16-31
    v8f acc10 = {};                  // edges 16-31, cols  0-15
    v8f acc11 = {};                  // edges 16-31, cols 16-31

    const float* fa0 = ft + mn * FSTRIDE + kb;          // A-tile 0
    const float* fa1 = ft + (16 + mn) * FSTRIDE + kb;   // A-tile 1
    const float* wb  = &W1f[lane * 2];

    #pragma unroll
    for (int ks = 0; ks < 20; ++ks) {
        const v2f a0 = *(const v2f*)(fa0 + ks * 4);
        const v2f a1 = *(const v2f*)(fa1 + ks * 4);
        const v2f w0 = *(const v2f*)(wb + ks * 128);
        const v2f w1 = *(const v2f*)(wb + ks * 128 + 64);
        // (neg_a, A, neg_b, B, c_mod, C, reuse_a, reuse_b)
        acc00 = __builtin_amdgcn_wmma_f32_16x16x4_f32(false, a0, false, w0,
                                                      (short)0, acc00, false, false);
        acc01 = __builtin_amdgcn_wmma_f32_16x16x4_f32(false, a0, false, w1,
                                                      (short)0, acc01, false, false);
        acc10 = __builtin_amdgcn_wmma_f32_16x16x4_f32(false, a1, false, w0,
                                                      (short)0, acc10, false, false);
        acc11 = __builtin_amdgcn_wmma_f32_16x16x4_f32(false, a1, false, w1,
                                                      (short)0, acc11, false, false);
    }

    // ---- Bias + ReLU + streaming store, fused row partial sums ----
    const float bia0 = b1s[mn];
    const float bia1 = b1s[16 + mn];
    float part[16];
    #pragma unroll
    for (int r = 0; r < 8; ++r) {
        const int m = r + half * 8;                 // row within each M sub-tile
        const float v0 = fmaxf(acc00[r] + bia0, 0.0f);
        const float v1 = fmaxf(acc01[r] + bia1, 0.0f);
        const float w0 = fmaxf(acc10[r] + bia0, 0.0f);
        const float w1 = fmaxf(acc11[r] + bia1, 0.0f);
        part[r]     = v0 + v1;
        part[8 + r] = w0 + w1;
        if (valid) {
            const size_t rb0 = ((size_t)b * E_ + (e0 + m))      * (size_t)EOUT_;
            const size_t rb1 = ((size_t)b * E_ + (e0 + 16 + m)) * (size_t)EOUT_;
            __builtin_nontemporal_store(v0, &new_e[rb0 + mn]);
            __builtin_nontemporal_store(v1, &new_e[rb0 + 16 + mn]);
            __builtin_nontemporal_store(w0, &new_e[rb1 + mn]);
            __builtin_nontemporal_store(w1, &new_e[rb1 + 16 + mn]);
        }
    }

    // ---- Transpose partials through LDS, reduce per-edge mean, scatter ----
    float* rd = &red[wave * 32 * 17];
    #pragma unroll
    for (int i = 0; i < 16; ++i) rd[lane * 17 + i] = part[i];
    asm volatile("s_wait_dscnt 0" ::: "memory");

    if (valid) {
        // Lane l reduces edge l of this tile (uses its own dstl).
        const int t  = lane >> 4;            // M sub-tile
        const int mm = lane & 15;            // row within sub-tile
        const int h  = mm >> 3;              // half whose lanes hold the partials
        const int r  = (mm & 7) + t * 8;     // part[] slot
        float tot = 0.0f;
        #pragma unroll
        for (int l = 0; l < 16; ++l) tot += rd[(h * 16 + l) * 17 + r];
        atomicAdd(&sums[(size_t)b * N_ + dstl], tot * (1.0f / (float)EOUT_));
    }
}

// ---------------------------------------------------------------------------
// Node MLP: new_x = relu([x | node_add] @ W2 + b2)
// One wave per (b, node); lane j computes output feature j.
// ---------------------------------------------------------------------------
__global__ __launch_bounds__(256) void node_mlp_kernel(
    const float* __restrict__ x,       // (B, N, 32)
    const float* __restrict__ W2,      // (33, 32) row-major
    const float* __restrict__ b2,      // (32,)
    const float* __restrict__ sums,    // (B, N)
    const float* __restrict__ counts,  // (N,)
    float*       __restrict__ new_x)   // (B, N, 32)
{
    const int gw   = (blockIdx.x * blockDim.x + threadIdx.x) >> 5;
    const int lane = threadIdx.x & 31;
    if (gw >= B_ * N_) return;
    const int b = gw / N_;
    const int n = gw - b * N_;

    const float cnt  = counts[n];
    const float s    = sums[(size_t)b * N_ + n];
    const float nadd = (cnt > 0.0f) ? (s / fmaxf(cnt, 1.0f)) : 0.0f;

    const size_t rb = ((size_t)b * N_ + n) * (size_t)XIN_;
    const float xv  = x[rb + lane];                        // lane holds x[b,n,lane]
    float acc = b2[lane] + nadd * W2[XIN_ * XOUT_ + lane]; // row 32 = node_add weight
    #pragma unroll
    for (int i = 0; i < XIN_; ++i)
        acc = fmaf(__shfl(xv, i, 32), W2[i * XOUT_ + lane], acc);
    new_x[rb + lane] = fmaxf(acc, 0.0f);
}

// ---------------------------------------------------------------------------
extern "C" void kernel_launch(void* const* d_in, const int* in_sizes, int n_in,
                              void* d_out, int out_size, void* d_ws, size_t ws_size,
                              hipStream_t stream) {
    const float* x     = (const float*)d_in[0];
    const float* e     = (const float*)d_in[1];
    const int*   edges = (const int*)  d_in[2];
    const float* W1    = (const float*)d_in[3];
    const float* b1    = (const float*)d_in[4];
    const float* W2    = (const float*)d_in[5];
    const float* b2    = (const float*)d_in[6];

    float* out   = (float*)d_out;
    float* new_x = out;                                   // (B,N,32) first
    float* new_e = out + (size_t)B_ * N_ * XOUT_;         // then (B,E,32)

    float* sums   = (float*)d_ws;                         // B*N floats
    float* counts = sums + (size_t)B_ * N_;               // N floats

    // 1) zero accumulators (graph-replay deterministic)
    const int ztot = B_ * N_ + N_;
    zero_ws_kernel<<<(ztot + 255) / 256, 256, 0, stream>>>(sums, ztot);

    // 2) per-node edge counts
    counts_kernel<<<(E_ + 255) / 256, 256, 0, stream>>>(edges, counts);

    // 3) edge MLP + fused mean scatter (WMMA f32 16x16x4, 80 per wave)
    dim3 egrid((E_ / 32 + 7) / 8, B_);
    edge_mlp_kernel<<<egrid, 256, 0, stream>>>(x, e, edges, W1, b1, new_e, sums);

    // 4) node MLP
    const int nthreads = B_ * N_ * 32;
    node_mlp_kernel<<<(nthreads + 255) / 256, 256, 0, stream>>>(x, W2, b2, sums,
                                                                counts, new_x);
}